// YoloLossV3_44676249813101
// MI455X (gfx1250) — compile-verified
//
#include <hip/hip_runtime.h>
#include <math.h>

// ---------------- problem constants (from setup_inputs) ----------------
#define NT   512     // targets
#define NB   16      // batch
#define NA   3       // anchors per level
#define NCLS 80
#define NO   85

// ---------------- workspace layout (int32 words) ----------------
#define WS_FLAGS 0        // uint bitmask, 403200 cells -> 12600 words
#define WS_NCNT  12600    // int[3] masked-target count per level
#define WS_POBJ  12608    // float[315] per-block obj partial sums
#define WS_SUMS  12928    // float[9]  (lvl*3 + {xy, wh, cls})
#define WS_ZEND  12944    // zero region end
#define WS_REC   13056    // 10 arrays x 1536 records (ints/floats)

typedef float v2f __attribute__((ext_vector_type(2)));
typedef float v8f __attribute__((ext_vector_type(8)));

__device__ __forceinline__ void levelDims(int lvl, int& h, int& cells, int& cbase) {
    if (lvl == 0)      { h = 80; cells = 307200; cbase = 0; }
    else if (lvl == 1) { h = 40; cells = 76800;  cbase = 307200; }
    else               { h = 20; cells = 19200;  cbase = 384000; }
}

__device__ __forceinline__ float splus(float x) {   // softplus, stable
    return log1pf(expf(-fabsf(x))) + fmaxf(x, 0.0f);
}
__device__ __forceinline__ float sigm(float x) { return 1.0f / (1.0f + expf(-x)); }

// Exact fp32 wave(32) sum using V_WMMA_F32_16X16X4_F32:
// A = ones(16x4), B holds the 32 lane partials in its 64 slots (other half 0).
// Each D column sums exactly 2 distinct lanes' values (the 32 lanes partition
// into the 16 columns), so 4 xor-folds over the 16 column sums finish the sum.
__device__ __forceinline__ float waveReduceSum(float v) {
    v2f a; a[0] = 1.0f; a[1] = 1.0f;
    v2f b; b[0] = v;    b[1] = 0.0f;
    v8f c = {};
    c = __builtin_amdgcn_wmma_f32_16x16x4_f32(
            /*neg_a=*/false, a, /*neg_b=*/false, b,
            /*c_mod=*/(short)0, c, /*reuse_a=*/false, /*reuse_b=*/false);
    float r = c[0];
    r += __shfl_xor(r, 1, 32);
    r += __shfl_xor(r, 2, 32);
    r += __shfl_xor(r, 4, 32);
    r += __shfl_xor(r, 8, 32);
    return r;   // full 32-lane sum on every lane
}

// Result valid on wave 0 (use tid==0). `red` must hold >= blockDim/32 floats.
__device__ __forceinline__ float blockReduceSum(float v, float* red) {
    int lane = threadIdx.x & 31;
    int wid  = threadIdx.x >> 5;
    int nw   = blockDim.x >> 5;
    v = waveReduceSum(v);
    __syncthreads();                 // protect red across repeated calls
    if (lane == 0) red[wid] = v;
    __syncthreads();
    float r = 0.0f;
    if (wid == 0) {                  // uniform per-wave branch: EXEC all-ones
        r = (lane < nw) ? red[lane] : 0.0f;
        r = waveReduceSum(r);
    }
    return r;
}

// ---------------- kernel 1: zero scratch accumulators ----------------
__global__ void k_zero(int* ws) {
    int i = blockIdx.x * blockDim.x + threadIdx.x;
    if (i < WS_ZEND) ws[i] = 0;
}

// ---------------- kernel 2: per-(level,target) prep + tobj scatter ----------------
__global__ void k_prep(const float* __restrict__ tgt, const float* __restrict__ anchors,
                       const float* __restrict__ imsz, int* ws) {
    int gid = blockIdx.x * blockDim.x + threadIdx.x;
    if (gid >= 3 * NT) return;
    int lvl = gid / NT, t = gid - lvl * NT;
    int h, cells, cbase; levelDims(lvl, h, cells, cbase);
    float sw = imsz[1] / (float)h;          // stride (w); square grid: w == h
    float sh = imsz[0] / (float)h;

    const float* tr = tgt + t * 6;
    int   b   = (int)tr[0];
    int   cls = (int)tr[1];
    float gx = tr[2] / sw, gy = tr[3] / sh;
    float gw = tr[4] / sw, gh = tr[5] / sh;

    float best = -1.0f, aw = 1.0f, ah = 1.0f; int abest = 0;
    for (int a = 0; a < NA; ++a) {
        float Aw = anchors[(lvl * NA + a) * 2 + 0] / sw;
        float Ah = anchors[(lvl * NA + a) * 2 + 1] / sh;
        float inter = fminf(Aw, gw) * fminf(Ah, gh);
        float uni   = Aw * Ah + gw * gh - inter;
        float iou   = inter / uni;
        if (iou > best) { best = iou; abest = a; aw = Aw; ah = Ah; }
    }
    int mask = (best > 0.2f) ? 1 : 0;
    int gi = (int)gx; gi = min(max(gi, 0), h - 1);
    int gj = (int)gy; gj = min(max(gj, 0), h - 1);

    int* rec = ws + WS_REC;
    float* recf = (float*)rec;
    int idx = lvl * NT + t;
    rec[0 * 1536 + idx] = abest;
    rec[1 * 1536 + idx] = b;
    rec[2 * 1536 + idx] = gi;
    rec[3 * 1536 + idx] = gj;
    rec[4 * 1536 + idx] = mask;
    rec[5 * 1536 + idx] = cls;
    recf[6 * 1536 + idx] = gx - floorf(gx);
    recf[7 * 1536 + idx] = gy - floorf(gy);
    recf[8 * 1536 + idx] = logf(gw / aw);
    recf[9 * 1536 + idx] = logf(gh / ah);

    if (mask) {
        atomicAdd(ws + WS_NCNT + lvl, 1);
        int cell = (((b * NA + abest) * h + gj) * h + gi) + cbase;
        atomicOr((unsigned int*)ws + (cell >> 5), 1u << (cell & 31));
    }
}

// ---------------- kernel 3: gathered lxy/lwh/lcls sums (1 block, 512 thr) ----------------
__global__ void k_gather(const float* __restrict__ p0, const float* __restrict__ p1,
                         const float* __restrict__ p2, int* ws) {
    __shared__ float red[16];
    int tid = threadIdx.x;
    const float* preds[3] = { p0, p1, p2 };
    int* rec = ws + WS_REC;
    float* recf = (float*)rec;
    float* sums = (float*)(ws + WS_SUMS);

    for (int lvl = 0; lvl < 3; ++lvl) {
        int h, cells, cb; levelDims(lvl, h, cells, cb);
        int idx = lvl * NT + tid;
        float sxy = 0.0f, swh = 0.0f, scls = 0.0f;
        if (rec[4 * 1536 + idx]) {
            int a  = rec[0 * 1536 + idx], b  = rec[1 * 1536 + idx];
            int gi = rec[2 * 1536 + idx], gj = rec[3 * 1536 + idx];
            int cls = rec[5 * 1536 + idx];
            const float* pi = preds[lvl] +
                (size_t)(((b * NA + a) * h + gj) * h + gi) * NO;
            float d0 = sigm(pi[0]) - recf[6 * 1536 + idx];
            float d1 = sigm(pi[1]) - recf[7 * 1536 + idx];
            sxy = d0 * d0 + d1 * d1;
            float e0 = pi[2] - recf[8 * 1536 + idx];
            float e1 = pi[3] - recf[9 * 1536 + idx];
            swh = e0 * e0 + e1 * e1;
            // bce(x, onehot) with pw=1 == sum softplus(x_c) - x_{cls-1}
            for (int c = 0; c < NCLS; ++c) scls += splus(pi[5 + c]);
            scls -= pi[5 + cls - 1];
        }
        float rs = blockReduceSum(sxy, red);
        if (tid == 0) sums[lvl * 3 + 0] = rs;
        rs = blockReduceSum(swh, red);
        if (tid == 0) sums[lvl * 3 + 1] = rs;
        rs = blockReduceSum(scls, red);
        if (tid == 0) sums[lvl * 3 + 2] = rs;
    }
}

// ---------------- kernel 4: objectness BCE over all cells ----------------
// blocks [0,240) -> lvl0, [240,300) -> lvl1, [300,315) -> lvl2; 1280 cells/block
__global__ void k_obj(const float* __restrict__ p0, const float* __restrict__ p1,
                      const float* __restrict__ p2, int* ws) {
    __shared__ float red[8];
    int blk = blockIdx.x;
    int lvl, lblk;
    if (blk < 240)      { lvl = 0; lblk = blk; }
    else if (blk < 300) { lvl = 1; lblk = blk - 240; }
    else                { lvl = 2; lblk = blk - 300; }
    int h, cells, cb; levelDims(lvl, h, cells, cb);
    const float* pi = (lvl == 0) ? p0 : ((lvl == 1) ? p1 : p2);
    const unsigned int* flags = (const unsigned int*)ws;

    int c0 = lblk * 1280;
    float s = 0.0f;
    for (int i = 0; i < 5; ++i) {
        int c = c0 + i * 256 + threadIdx.x;               // always < cells
        __builtin_prefetch(&pi[(size_t)(c + 1280) * NO + 4], 0, 0);
        float x = pi[(size_t)c * NO + 4];
        unsigned int f = (flags[(cb + c) >> 5] >> ((cb + c) & 31)) & 1u;
        // bce(x,t) pw=1:  t=1 -> softplus(-x),  t=0 -> softplus(x)
        s += f ? splus(-x) : splus(x);
    }
    float rs = blockReduceSum(s, red);
    if (threadIdx.x == 0) ((float*)(ws + WS_POBJ))[blk] = rs;
}

// ---------------- kernel 5: final combine ----------------
__global__ void k_combine(int* ws, float* out) {
    __shared__ float red[16];
    __shared__ float lob[3];
    int tid = threadIdx.x;
    const float* pobj = (const float*)(ws + WS_POBJ);
    const int   pb[4]     = { 0, 240, 300, 315 };
    const float cellsF[3] = { 307200.0f, 76800.0f, 19200.0f };

    for (int lvl = 0; lvl < 3; ++lvl) {
        float s = 0.0f;
        for (int i = pb[lvl] + tid; i < pb[lvl + 1]; i += blockDim.x) s += pobj[i];
        float rs = blockReduceSum(s, red);
        if (tid == 0) lob[lvl] = rs / cellsF[lvl];
        __syncthreads();
    }
    if (tid == 0) {
        const float* sums = (const float*)(ws + WS_SUMS);
        const int*   ncnt = ws + WS_NCNT;
        float lxy = 0.0f, lwh = 0.0f, lobj = 0.0f, lcls = 0.0f;
        for (int lvl = 0; lvl < 3; ++lvl) {
            float n = fmaxf((float)ncnt[lvl], 1.0f);
            lxy  += sums[lvl * 3 + 0] / (n * 2.0f);
            lwh  += sums[lvl * 3 + 1] / (n * 2.0f);
            lcls += sums[lvl * 3 + 2] / (n * (float)NCLS);
            lobj += lob[lvl];
        }
        lxy *= 0.05f; lwh *= 0.05f; lcls *= 0.5f;   // HYP weights; obj = 1.0
        float total = (lxy + lwh + lobj + lcls) * (float)NB;
        out[0] = total; out[1] = lxy; out[2] = lwh; out[3] = lobj; out[4] = lcls;
    }
}

extern "C" void kernel_launch(void* const* d_in, const int* in_sizes, int n_in,
                              void* d_out, int out_size, void* d_ws, size_t ws_size,
                              hipStream_t stream) {
    const float* p0      = (const float*)d_in[0];
    const float* p1      = (const float*)d_in[1];
    const float* p2      = (const float*)d_in[2];
    const float* targets = (const float*)d_in[3];
    const float* anchors = (const float*)d_in[4];
    const float* imsz    = (const float*)d_in[5];
    int*   ws  = (int*)d_ws;
    float* out = (float*)d_out;

    k_zero   <<<51, 256, 0, stream>>>(ws);
    k_prep   <<<6, 256, 0, stream>>>(targets, anchors, imsz, ws);
    k_gather <<<1, 512, 0, stream>>>(p0, p1, p2, ws);
    k_obj    <<<315, 256, 0, stream>>>(p0, p1, p2, ws);
    k_combine<<<1, 512, 0, stream>>>(ws, out);
}